// AdaptiveDiffusionBlock_83330955477705
// MI455X (gfx1250) — compile-verified
//
#include <hip/hip_runtime.h>
#include <math.h>

// ---------------------------------------------------------------------------
// AdaptiveDiffusionBlock on MI455X (gfx1250).
//   - fp32 matmuls via V_WMMA_F32_16X16X4_F32
//   - GEMM tile staging via Tensor Data Mover (tensor_load_to_lds) with
//     double buffering, hardware LDS padding, and s_wait_tensorcnt.
// ---------------------------------------------------------------------------

typedef float v2f __attribute__((ext_vector_type(2)));
typedef float v8f __attribute__((ext_vector_type(8)));

#define NP   2048
#define NC   64
#define CC   128
#define DD   64
#define P_TOPP 0.9f

#define GBM 128
#define GBN 128
#define GBK 32
#define ASTR 36    // A tile LDS row stride (dwords): 32 + 4 pad
#define BSTR 132   // B tile LDS row stride (dwords): 128 + 4 pad
#define ABUF (GBM * ASTR)   // 4608 floats
#define BBUF (GBK * BSTR)   // 4224 floats

#if __has_builtin(__builtin_amdgcn_tensor_load_to_lds) && \
    __has_builtin(__builtin_amdgcn_s_wait_tensorcnt)
#define USE_TDM 1
#else
#define USE_TDM 0
#endif

#if USE_TDM
typedef unsigned int u32x4 __attribute__((ext_vector_type(4)));
typedef int i32x4 __attribute__((ext_vector_type(4)));
typedef int i32x8 __attribute__((ext_vector_type(8)));

// Issue one 2D TDM tile load Global->LDS (6-arg clang-23 builtin form).
//   tile_w (dwords per row), tile_h (rows), row_stride (elements in global),
//   pad codes per D# spec: interval 2^(c+1) dwords, amount (c+1) dwords.
__device__ __forceinline__ void tdm_load_tile(const float* gptr, float* lptr,
                                              int tile_w, int tile_h, int row_stride,
                                              int pad_int_code, int pad_amt_code)
{
  unsigned long long ga = (unsigned long long)(uintptr_t)gptr;
  u32x4 g0;
  g0.x = 1u;                                     // count=1 (valid), user mode
  g0.y = (unsigned int)(uintptr_t)lptr;          // LDS byte address (addr[31:0])
  g0.z = (unsigned int)ga;                       // global_addr[31:0]
  g0.w = (unsigned int)(ga >> 32) | (2u << 30);  // global_addr[56:32] | type=2
  i32x8 g1;
  g1[0] = (2 << 16)                              // data_size = 4B
        | (1 << 20)                              // pad_enable
        | (pad_int_code << 22) | (pad_amt_code << 25);
  g1[1] = (int)((unsigned)tile_w << 16);         // tensor_dim0[15:0] (== tile_w)
  g1[2] = (int)(((unsigned)tile_w >> 16) | ((unsigned)tile_h << 16)); // dim0 hi | dim1 lo
  g1[3] = (int)(((unsigned)tile_h >> 16) | ((unsigned)tile_w << 16)); // dim1 hi | tile_dim0
  g1[4] = tile_h & 0xFFFF;                       // tile_dim1
  g1[5] = row_stride;                            // tensor_dim0_stride[31:0]
  g1[6] = 0;                                     // stride hi | dim1_stride lo
  g1[7] = 0;
  i32x4 gz4 = {0, 0, 0, 0};                      // 2D tensor: groups 2/3 unused
  i32x8 gz8 = {0, 0, 0, 0, 0, 0, 0, 0};
  __builtin_amdgcn_tensor_load_to_lds(g0, g1, gz4, gz4, gz8, 0);
}
#endif

// -------- one 128x128x32 WMMA macro-step; As m-major(36), Bs k-major(132) ---
__device__ __forceinline__ void wmma_chunk(const float* As, const float* Bs,
                                           int wr, int wc, int lh, int lm,
                                           v8f acc[4][2])
{
  #pragma unroll
  for (int kk = 0; kk < GBK; kk += 4) {
    v2f af[4], bf[2];
    #pragma unroll
    for (int i = 0; i < 4; ++i) {
      int m = (wr * 4 + i) * 16 + lm;            // A frag: lane = M, half picks K pair
      af[i] = *(const v2f*)(As + m * ASTR + kk + lh * 2);
    }
    #pragma unroll
    for (int j = 0; j < 2; ++j) {
      int n = (wc * 2 + j) * 16 + lm;            // B frag: lane = N
      bf[j].x = Bs[(kk + lh * 2 + 0) * BSTR + n];
      bf[j].y = Bs[(kk + lh * 2 + 1) * BSTR + n];
    }
    #pragma unroll
    for (int i = 0; i < 4; ++i)
      #pragma unroll
      for (int j = 0; j < 2; ++j)
        acc[i][j] = __builtin_amdgcn_wmma_f32_16x16x4_f32(
            false, af[i], false, bf[j], (short)0, acc[i][j], false, false);
  }
}

// ------------------------- tiled WMMA f32 GEMM -----------------------------
// C[M,N] = scale * A[M,K] @ B  (+ C if accum)
//   BT=false: B row-major [K,N] (TDM-staged, double buffered)
//   BT=true : B row-major [N,K] (multiply by B^T; manual staging)
template <bool BT>
__global__ void __launch_bounds__(256) gemm_wmma(
    float* __restrict__ C, const float* __restrict__ A, const float* __restrict__ B,
    int M, int N, int K, int lda, int ldb, int ldc, float scale, int accum)
{
  constexpr int NBUF = ((USE_TDM != 0) && !BT) ? 2 : 1;
  __shared__ float As[NBUF * ABUF];
  __shared__ float Bs[NBUF * BBUF];

  const int m0   = blockIdx.y * GBM;
  const int n0   = blockIdx.x * GBN;
  const int tid  = threadIdx.x;
  const int lane = tid & 31;
  const int wave = tid >> 5;          // 8 waves
  const int wr   = wave >> 2;         // 0..1
  const int wc   = wave & 3;          // 0..3
  const int lh   = lane >> 4;
  const int lm   = lane & 15;

  v8f acc[4][2];
  const v8f z = {0.f,0.f,0.f,0.f,0.f,0.f,0.f,0.f};
  for (int i = 0; i < 4; ++i)
    for (int j = 0; j < 2; ++j) acc[i][j] = z;

#if USE_TDM
  if constexpr (!BT) {
    // -------- TDM double-buffered pipeline --------
    const int nch = K / GBK;
    if (wave == 0) {
      tdm_load_tile(A + (size_t)m0 * lda, As, GBK, GBM, lda, 4, 3);          // A chunk 0
      tdm_load_tile(B + (size_t)0 * ldb + n0, Bs, GBN, GBK, ldb, 6, 3);      // B chunk 0
    }
    for (int i = 0; i < nch; ++i) {
      const int cur = i & 1;
      if (wave == 0) {
        if (i + 1 < nch) {
          const int nxt = cur ^ 1;
          const int k0n = (i + 1) * GBK;
          tdm_load_tile(A + (size_t)m0 * lda + k0n, As + nxt * ABUF, GBK, GBM, lda, 4, 3);
          tdm_load_tile(B + (size_t)k0n * ldb + n0, Bs + nxt * BBUF, GBN, GBK, ldb, 6, 3);
          __builtin_amdgcn_s_wait_tensorcnt(2);   // chunk i complete (in-order TDM)
        } else {
          __builtin_amdgcn_s_wait_tensorcnt(0);
        }
      }
      __syncthreads();                            // publish chunk i to all waves
      wmma_chunk(As + cur * ABUF, Bs + cur * BBUF, wr, wc, lh, lm, acc);
      __syncthreads();                            // done reading buf before reuse
    }
  } else
#endif
  {
    // -------- manual single-buffered staging --------
    for (int k0 = 0; k0 < K; k0 += GBK) {
      for (int it = 0; it < 4; ++it) {            // A tile: 1024 float4 units
        int q  = tid + it * 256;
        int k4 = q & 7;
        int m  = q >> 3;
        const float4 v = *(const float4*)(A + (size_t)(m0 + m) * lda + k0 + k4 * 4);
        *(float4*)(&As[m * ASTR + k4 * 4]) = v;
      }
      if constexpr (!BT) {
        for (int it = 0; it < 4; ++it) {
          int q  = tid + it * 256;
          int n4 = q & 31;
          int k  = q >> 5;
          const float4 v = *(const float4*)(B + (size_t)(k0 + k) * ldb + n0 + n4 * 4);
          *(float4*)(&Bs[k * BSTR + n4 * 4]) = v;
        }
      } else {
        for (int it = 0; it < 4; ++it) {
          int q  = tid + it * 256;
          int k4 = q & 7;
          int n  = q >> 3;
          const float4 v = *(const float4*)(B + (size_t)(n0 + n) * ldb + k0 + k4 * 4);
          Bs[(k4*4+0)*BSTR + n] = v.x;
          Bs[(k4*4+1)*BSTR + n] = v.y;
          Bs[(k4*4+2)*BSTR + n] = v.z;
          Bs[(k4*4+3)*BSTR + n] = v.w;
        }
      }
      __syncthreads();
      wmma_chunk(As, Bs, wr, wc, lh, lm, acc);
      __syncthreads();
    }
  }

  // ---- epilogue: C/D frag layout -> row = r + 8*half, col = lm
  for (int i = 0; i < 4; ++i) {
    int gmb = m0 + (wr * 4 + i) * 16 + lh * 8;
    for (int j = 0; j < 2; ++j) {
      int gn = n0 + (wc * 2 + j) * 16 + lm;
      union { v8f v; float f[8]; } u; u.v = acc[i][j];
      #pragma unroll
      for (int r = 0; r < 8; ++r) {
        size_t o = (size_t)(gmb + r) * ldc + gn;
        float val = u.f[r] * scale;
        if (accum) val += C[o];
        C[o] = val;
      }
    }
  }
}

// ------------------------- row-normalised prior ----------------------------
__global__ void __launch_bounds__(256) rownorm_kernel(float* __restrict__ Rf,
                                                      const float* __restrict__ prior)
{
  __shared__ float red[256];
  const int i = blockIdx.x, t = threadIdx.x;
  const float* r = prior + (size_t)i * NP;
  float s = 0.f;
  for (int j = t; j < NP; j += 256) s += r[j];
  red[t] = s; __syncthreads();
  for (int o = 128; o > 0; o >>= 1) { if (t < o) red[t] += red[t + o]; __syncthreads(); }
  const float inv = 1.0f / fmaxf(red[0], 1e-12f);
  for (int j = t; j < NP; j += 256) Rf[(size_t)i * NP + j] = r[j] * inv;
}

// ------------------------- mean-pool over protein axis ---------------------
__global__ void pool_mean_kernel(float* __restrict__ pooled, const float* __restrict__ X)
{
  const int i = blockIdx.x, c = threadIdx.x;           // 128 threads
  const float* base = X + (size_t)i * NC * CC + c;
  float s = 0.f;
  for (int k = 0; k < NC; ++k) s += base[(size_t)k * CC];
  pooled[(size_t)i * CC + c] = s * (1.0f / (float)NC);
}

// ------------------------- tiny GEMM: out[n][m] = in[n][:]·W[m][:] ---------
__global__ void small_gemm64_kernel(float* __restrict__ out, const float* __restrict__ in,
                                    const float* __restrict__ W, int K)
{
  __shared__ float row[CC];
  const int n = blockIdx.x, m = threadIdx.x;           // 64 threads
  for (int k = m; k < K; k += 64) row[k] = in[(size_t)n * K + k];
  __syncthreads();
  float a = 0.f;
  for (int k = 0; k < K; ++k) a += row[k] * W[(size_t)m * K + k];
  out[(size_t)n * 64 + m] = a;
}

// ----------------- softmax + nucleus (top-p) mask + renorm -----------------
__global__ void __launch_bounds__(256) softmax_topp_kernel(float* __restrict__ attn)
{
  __shared__ float p[NP];      // probs (original order)
  __shared__ float s[NP];      // sorted copy (descending)
  __shared__ float red[256];
  const int row = blockIdx.x, t = threadIdx.x;
  float* g = attn + (size_t)row * NP;

  float lmax = -INFINITY;
  for (int j = t; j < NP; j += 256) { float v = g[j]; p[j] = v; lmax = fmaxf(lmax, v); }
  red[t] = lmax; __syncthreads();
  for (int o = 128; o > 0; o >>= 1) { if (t < o) red[t] = fmaxf(red[t], red[t + o]); __syncthreads(); }
  const float rmax = red[0]; __syncthreads();

  float lsum = 0.f;
  for (int j = t; j < NP; j += 256) { float e = expf(p[j] - rmax); p[j] = e; lsum += e; }
  red[t] = lsum; __syncthreads();
  for (int o = 128; o > 0; o >>= 1) { if (t < o) red[t] += red[t + o]; __syncthreads(); }
  const float inv = 1.0f / red[0]; __syncthreads();
  for (int j = t; j < NP; j += 256) { float pr = p[j] * inv; p[j] = pr; s[j] = pr; }
  __syncthreads();

  // bitonic sort s[] descending
  for (int ksz = 2; ksz <= NP; ksz <<= 1) {
    for (int j = ksz >> 1; j > 0; j >>= 1) {
      for (int a = t; a < NP; a += 256) {
        int b = a ^ j;
        if (b > a) {
          float va = s[a], vb = s[b];
          bool asc = (a & ksz) != 0;
          if (asc ? (va > vb) : (va < vb)) { s[a] = vb; s[b] = va; }
        }
      }
      __syncthreads();
    }
  }

  // exclusive prefix over 8-element chunks -> nucleus threshold
  float chunk = 0.f;
  for (int q = 0; q < 8; ++q) chunk += s[t * 8 + q];
  red[t] = chunk; __syncthreads();
  for (int off = 1; off < 256; off <<= 1) {
    float v = red[t];
    float add = (t >= off) ? red[t - off] : 0.f;
    __syncthreads();
    red[t] = v + add;
    __syncthreads();
  }
  float run = (t == 0) ? 0.f : red[t - 1];
  float cand = INFINITY;
  for (int q = 0; q < 8; ++q) {
    float v = s[t * 8 + q];
    if (run < P_TOPP) cand = v;
    run += v;
  }
  __syncthreads();
  red[t] = cand; __syncthreads();
  for (int o = 128; o > 0; o >>= 1) { if (t < o) red[t] = fminf(red[t], red[t + o]); __syncthreads(); }
  const float thr = red[0]; __syncthreads();

  float msum = 0.f;
  for (int j = t; j < NP; j += 256) { float v = p[j]; if (v < thr) v = 0.f; p[j] = v; msum += v; }
  red[t] = msum; __syncthreads();
  for (int o = 128; o > 0; o >>= 1) { if (t < o) red[t] += red[t + o]; __syncthreads(); }
  const float minv = 1.0f / fmaxf(red[0], 1e-12f);
  for (int j = t; j < NP; j += 256) g[j] = p[j] * minv;
}

// ----------------------- out = LN(x + gate*residual) -----------------------
__global__ void finalize_ln_kernel(float* __restrict__ out, const float* __restrict__ x,
                                   const float* __restrict__ gamma, const float* __restrict__ beta,
                                   const float* __restrict__ alpha)
{
  __shared__ float red[128];
  const int row = blockIdx.x, c = threadIdx.x;       // 128 threads
  const size_t idx = (size_t)row * CC + c;
  const float gate = 1.0f / (1.0f + expf(-alpha[0]));
  const float h = x[idx] + gate * out[idx];
  red[c] = h; __syncthreads();
  for (int o = 64; o > 0; o >>= 1) { if (c < o) red[c] += red[c + o]; __syncthreads(); }
  const float mu = red[0] * (1.0f / (float)CC); __syncthreads();
  const float d = h - mu;
  red[c] = d * d; __syncthreads();
  for (int o = 64; o > 0; o >>= 1) { if (c < o) red[c] += red[c + o]; __syncthreads(); }
  const float var = red[0] * (1.0f / (float)CC);
  out[idx] = d * rsqrtf(var + 1e-5f) * gamma[c] + beta[c];
}

// ---------------------------------------------------------------------------
extern "C" void kernel_launch(void* const* d_in, const int* in_sizes, int n_in,
                              void* d_out, int out_size, void* d_ws, size_t ws_size,
                              hipStream_t stream)
{
  (void)in_sizes; (void)n_in; (void)out_size; (void)ws_size;

  const float* x     = (const float*)d_in[0];   // (2048,64,128)
  const float* prior = (const float*)d_in[1];   // (2048,2048)
  const float* W1    = (const float*)d_in[2];   // (64,128)
  const float* W2    = (const float*)d_in[3];   // (64,64)
  const float* W3    = (const float*)d_in[4];   // (64,128)
  const float* Wf    = (const float*)d_in[5];   // (2,128,128)
  const float* Wa    = (const float*)d_in[6];   // (2,128,128)
  const float* gamma = (const float*)d_in[7];
  const float* beta  = (const float*)d_in[8];
  const float* alpha = (const float*)d_in[9];
  float* res = (float*)d_out;                   // residual accumulator -> final output

  const size_t NPNP = (size_t)NP * NP;          // 4M floats
  const size_t XSZ  = (size_t)NP * NC * CC;     // 16.7M floats

  float* w      = (float*)d_ws;
  float* Rf     = w; w += NPNP;
  float* attn   = w; w += NPNP;
  float* pooled = w; w += (size_t)NP * CC;
  float* e1     = w; w += (size_t)NP * DD;
  float* e3     = w; w += (size_t)NP * DD;
  float* tbuf   = w; w += (size_t)NP * DD;
  float* XF     = w; w += XSZ;
  float* XA     = w; w += XSZ;
  float* T      = w; w += XSZ;

  rownorm_kernel<<<NP, 256, 0, stream>>>(Rf, prior);

  for (int step = 0; step < 2; ++step) {
    const float* Xa_src = (step == 0) ? x : XA;
    const float* Xf_src = (step == 0) ? x : XF;

    pool_mean_kernel<<<NP, 128, 0, stream>>>(pooled, Xa_src);
    small_gemm64_kernel<<<NP, 64, 0, stream>>>(e1, pooled, W1, CC);
    small_gemm64_kernel<<<NP, 64, 0, stream>>>(e3, pooled, W3, CC);
    small_gemm64_kernel<<<NP, 64, 0, stream>>>(tbuf, e1, W2, DD);

    // logits = (e1@W2^T)@e3^T / TAU   (TAU = 1)
    gemm_wmma<true><<<dim3(NP / GBN, NP / GBM), 256, 0, stream>>>(
        attn, tbuf, e3, NP, NP, DD, DD, DD, NP, 1.0f, 0);
    softmax_topp_kernel<<<NP, 256, 0, stream>>>(attn);

    // Xf_new = Rf @ Xf ; residual += Xf_new @ Wf[step]^T
    float* Fdst = (step == 0) ? XF : T;
    gemm_wmma<false><<<dim3((NC * CC) / GBN, NP / GBM), 256, 0, stream>>>(
        Fdst, Rf, Xf_src, NP, NC * CC, NP, NP, NC * CC, NC * CC, 1.0f, 0);
    gemm_wmma<true><<<dim3(1, (NP * NC) / GBM), 256, 0, stream>>>(
        res, Fdst, Wf + (size_t)step * CC * CC, NP * NC, CC, CC, CC, CC, CC,
        1.0f, (step == 0) ? 0 : 1);

    // Xa_new = attn @ Xa ; residual += Xa_new @ Wa[step]^T
    float* Adst = (step == 0) ? XA : T;
    gemm_wmma<false><<<dim3((NC * CC) / GBN, NP / GBM), 256, 0, stream>>>(
        Adst, attn, Xa_src, NP, NC * CC, NP, NP, NC * CC, NC * CC, 1.0f, 0);
    gemm_wmma<true><<<dim3(1, (NP * NC) / GBM), 256, 0, stream>>>(
        res, Adst, Wa + (size_t)step * CC * CC, NP * NC, CC, CC, CC, CC, CC,
        1.0f, 1);
  }

  finalize_ln_kernel<<<NP * NC, 128, 0, stream>>>(res, x, gamma, beta, alpha);
}